// GCN_47605417508956
// MI455X (gfx1250) — compile-verified
//
#include <hip/hip_runtime.h>

typedef __attribute__((ext_vector_type(2))) float v2f;
typedef __attribute__((ext_vector_type(8))) float v8f;

#define NFEAT 128
#define NHID  64
#define NCLASS 32

// ---------------------------------------------------------------------------
// init: deg = 1.0 (self-loop), acc1 = 0
__global__ __launch_bounds__(256) void k_init(float* __restrict__ deg,
                                              float* __restrict__ acc1,
                                              long long n) {
    size_t i = (size_t)blockIdx.x * blockDim.x + threadIdx.x;
    size_t tot = (size_t)n * NHID;
    if (i < tot) acc1[i] = 0.0f;
    if (i < (size_t)n) deg[i] = 1.0f;
}

__global__ __launch_bounds__(256) void k_zero(float* __restrict__ p, long long n) {
    size_t i = (size_t)blockIdx.x * blockDim.x + threadIdx.x;
    if (i < (size_t)n) p[i] = 0.0f;
}

// degree from in-edges
__global__ __launch_bounds__(256) void k_deg(const long long* __restrict__ dst,
                                             float* __restrict__ deg, long long E) {
    size_t i = (size_t)blockIdx.x * blockDim.x + threadIdx.x;
    if (i < (size_t)E) atomicAdd(&deg[(size_t)dst[i]], 1.0f);
}

__global__ __launch_bounds__(256) void k_dinv(const float* __restrict__ deg,
                                              float* __restrict__ dinv, long long n) {
    size_t i = (size_t)blockIdx.x * blockDim.x + threadIdx.x;
    if (i < (size_t)n) dinv[i] = rsqrtf(deg[i]);   // deg >= 1 always (self-loop)
}

// ---------------------------------------------------------------------------
// GEMM1: hs[row, 0:64] = (x[row, 0:128] @ w1[128,64]) * dinv[row]
// One wave per 16-row tile; 4 column tiles of 16; K stepped by 4 via
// v_wmma_f32_16x16x4_f32.  W1 staged in LDS pair-interleaved:
//   w1s[(k>>1)*2*NHID + col*2 + (k&1)] = w1[k*NHID + col]
// so each B fragment (w[kk][col], w[kk+1][col]) is one contiguous b64 read.
__global__ __launch_bounds__(256) void k_gemm1(const float* __restrict__ x,
                                               const float* __restrict__ w1,
                                               const float* __restrict__ dinv,
                                               float* __restrict__ hs,
                                               int nrows) {
    __shared__ float w1s[NFEAT * NHID];
    for (int i = threadIdx.x; i < NFEAT * NHID; i += 256) {
        int k = i / NHID, col = i % NHID;
        w1s[(k >> 1) * (2 * NHID) + col * 2 + (k & 1)] = w1[i];
    }
    __syncthreads();

    const int wave = threadIdx.x >> 5;
    const int lane = threadIdx.x & 31;
    const int rowBase = (blockIdx.x * 8 + wave) * 16;
    if (rowBase >= nrows) return;            // uniform per wave: EXEC stays all-1s

    const int lm = lane & 15;
    const int kh = lane >> 4;                // 0 or 1: which K-half this lane holds

    int arow = rowBase + lm;                 // A-fragment row for this lane
    if (arow >= nrows) arow = nrows - 1;     // clamp (keeps EXEC full)

    v8f c0 = {}, c1 = {}, c2 = {}, c3 = {};
    for (int k0 = 0; k0 < NFEAT; k0 += 4) {
        const int kk = k0 + 2 * kh;          // even; lane holds (kk, kk+1)
        const float* xp = x + (size_t)arow * NFEAT + kk;
        v2f a; a.x = xp[0]; a.y = xp[1];

        // pair-interleaved: fragment base = (kk>>1)*2*NHID + col*2
        const float* wp = w1s + (kk >> 1) * (2 * NHID) + lm * 2;
        const v2f b0 = *(const v2f*)(wp + 0);
        const v2f b1 = *(const v2f*)(wp + 32);
        const v2f b2 = *(const v2f*)(wp + 64);
        const v2f b3 = *(const v2f*)(wp + 96);

        c0 = __builtin_amdgcn_wmma_f32_16x16x4_f32(false, a, false, b0, (short)0, c0, false, false);
        c1 = __builtin_amdgcn_wmma_f32_16x16x4_f32(false, a, false, b1, (short)0, c1, false, false);
        c2 = __builtin_amdgcn_wmma_f32_16x16x4_f32(false, a, false, b2, (short)0, c2, false, false);
        c3 = __builtin_amdgcn_wmma_f32_16x16x4_f32(false, a, false, b3, (short)0, c3, false, false);
    }

    // C layout: VGPR r -> row = rowBase + r + 8*kh, col = tile*16 + lm
#pragma unroll
    for (int r = 0; r < 8; ++r) {
        const int row = rowBase + r + 8 * kh;
        if (row >= nrows) continue;
        const float s = dinv[row];
        float* o = hs + (size_t)row * NHID + lm;
        o[0]  = c0[r] * s;
        o[16] = c1[r] * s;
        o[32] = c2[r] * s;
        o[48] = c3[r] * s;
    }
}

// GEMM2: gs[row, 0:32] = (h1[row, 0:64] @ w2[64,32]) * dinv[row]
__global__ __launch_bounds__(256) void k_gemm2(const float* __restrict__ h1,
                                               const float* __restrict__ w2,
                                               const float* __restrict__ dinv,
                                               float* __restrict__ gs,
                                               int nrows) {
    __shared__ float w2s[NHID * NCLASS];
    for (int i = threadIdx.x; i < NHID * NCLASS; i += 256) {
        int k = i / NCLASS, col = i % NCLASS;
        w2s[(k >> 1) * (2 * NCLASS) + col * 2 + (k & 1)] = w2[i];
    }
    __syncthreads();

    const int wave = threadIdx.x >> 5;
    const int lane = threadIdx.x & 31;
    const int rowBase = (blockIdx.x * 8 + wave) * 16;
    if (rowBase >= nrows) return;

    const int lm = lane & 15;
    const int kh = lane >> 4;

    int arow = rowBase + lm;
    if (arow >= nrows) arow = nrows - 1;

    v8f c0 = {}, c1 = {};
    for (int k0 = 0; k0 < NHID; k0 += 4) {
        const int kk = k0 + 2 * kh;
        const float* hp = h1 + (size_t)arow * NHID + kk;
        v2f a; a.x = hp[0]; a.y = hp[1];

        const float* wp = w2s + (kk >> 1) * (2 * NCLASS) + lm * 2;
        const v2f b0 = *(const v2f*)(wp + 0);
        const v2f b1 = *(const v2f*)(wp + 32);

        c0 = __builtin_amdgcn_wmma_f32_16x16x4_f32(false, a, false, b0, (short)0, c0, false, false);
        c1 = __builtin_amdgcn_wmma_f32_16x16x4_f32(false, a, false, b1, (short)0, c1, false, false);
    }

#pragma unroll
    for (int r = 0; r < 8; ++r) {
        const int row = rowBase + r + 8 * kh;
        if (row >= nrows) continue;
        const float s = dinv[row];
        float* o = gs + (size_t)row * NCLASS + lm;
        o[0]  = c0[r] * s;
        o[16] = c1[r] * s;
    }
}

// ---------------------------------------------------------------------------
// scatter-add: acc[dst, :] += hs[src, :]   (64 features, 16 threads/edge, float4)
__global__ __launch_bounds__(256) void k_scat64(const long long* __restrict__ src,
                                                const long long* __restrict__ dst,
                                                const float* __restrict__ hs,
                                                float* __restrict__ acc,
                                                long long E) {
    size_t idx = (size_t)blockIdx.x * blockDim.x + threadIdx.x;
    size_t tot = (size_t)E * 16;
    if (idx >= tot) return;
    size_t e = idx >> 4;
    int f = (int)(idx & 15) * 4;
    size_t s = (size_t)src[e];
    size_t d = (size_t)dst[e];
    const float4 v = *(const float4*)(hs + s * NHID + f);
    float* o = acc + d * NHID + f;
    atomicAdd(o + 0, v.x);
    atomicAdd(o + 1, v.y);
    atomicAdd(o + 2, v.z);
    atomicAdd(o + 3, v.w);
}

// scatter-add: acc[dst, :] += gs[src, :]   (32 features, 8 threads/edge, float4)
__global__ __launch_bounds__(256) void k_scat32(const long long* __restrict__ src,
                                                const long long* __restrict__ dst,
                                                const float* __restrict__ gs,
                                                float* __restrict__ acc,
                                                long long E) {
    size_t idx = (size_t)blockIdx.x * blockDim.x + threadIdx.x;
    size_t tot = (size_t)E * 8;
    if (idx >= tot) return;
    size_t e = idx >> 3;
    int f = (int)(idx & 7) * 4;
    size_t s = (size_t)src[e];
    size_t d = (size_t)dst[e];
    const float4 v = *(const float4*)(gs + s * NCLASS + f);
    float* o = acc + d * NCLASS + f;
    atomicAdd(o + 0, v.x);
    atomicAdd(o + 1, v.y);
    atomicAdd(o + 2, v.z);
    atomicAdd(o + 3, v.w);
}

// ---------------------------------------------------------------------------
// finalize layer1 in-place: acc1 <- relu(dinv[i]*(acc1 + hs) + b1[f])
__global__ __launch_bounds__(256) void k_fin1(float* __restrict__ acc1,
                                              const float* __restrict__ hs,
                                              const float* __restrict__ dinv,
                                              const float* __restrict__ b1,
                                              long long n) {
    size_t idx = (size_t)blockIdx.x * blockDim.x + threadIdx.x;
    size_t tot = (size_t)n * NHID;
    if (idx >= tot) return;
    size_t i = idx >> 6;           // / NHID
    int f = (int)(idx & (NHID - 1));
    float v = dinv[i] * (acc1[idx] + hs[idx]) + b1[f];
    acc1[idx] = fmaxf(v, 0.0f);
}

// finalize layer2 to output: out = dinv[i]*(acc2 + gs) + b2[f]
__global__ __launch_bounds__(256) void k_fin2(const float* __restrict__ acc2,
                                              const float* __restrict__ gs,
                                              const float* __restrict__ dinv,
                                              const float* __restrict__ b2,
                                              float* __restrict__ out,
                                              long long n) {
    size_t idx = (size_t)blockIdx.x * blockDim.x + threadIdx.x;
    size_t tot = (size_t)n * NCLASS;
    if (idx >= tot) return;
    size_t i = idx >> 5;           // / NCLASS
    int f = (int)(idx & (NCLASS - 1));
    out[idx] = dinv[i] * (acc2[idx] + gs[idx]) + b2[f];
}

// ---------------------------------------------------------------------------
extern "C" void kernel_launch(void* const* d_in, const int* in_sizes, int n_in,
                              void* d_out, int out_size, void* d_ws, size_t ws_size,
                              hipStream_t stream) {
    const float* x  = (const float*)d_in[0];
    const float* w1 = (const float*)d_in[1];
    const float* b1 = (const float*)d_in[2];
    const float* w2 = (const float*)d_in[3];
    const float* b2 = (const float*)d_in[4];
    const long long* ei = (const long long*)d_in[5];   // int64 [2, E]

    const long long N = (long long)in_sizes[0] / NFEAT;
    const long long E = (long long)in_sizes[5] / 2;
    const long long* srcp = ei;       // edge_index[0]
    const long long* dstp = ei + E;   // edge_index[1]

    // workspace layout (floats): deg[N] | dinv[N] | acc1[64N] | hs1[64N]
    // gs2 (32N) and acc2 (32N) overlay hs1 after it is dead.
    float* ws   = (float*)d_ws;
    float* deg  = ws;
    float* dinv = ws + N;
    float* acc1 = ws + 2 * N;
    float* hs1  = ws + 2 * N + 64 * N;
    float* gs2  = hs1;                 // reuse: written by gemm2 after fin1
    float* acc2 = hs1 + 32 * N;        // reuse: zeroed after fin1
    float* out  = (float*)d_out;

    const int B = 256;
    auto blk = [](long long t, int b) { return (unsigned)((t + b - 1) / b); };

    const long long tiles  = (N + 15) / 16;
    const unsigned  gblk   = (unsigned)((tiles + 7) / 8);   // 8 waves (tiles) per block

    // 1) deg=1 (self-loop), acc1=0
    k_init<<<blk(N * NHID, B), B, 0, stream>>>(deg, acc1, N);
    // 2) in-degree
    k_deg<<<blk(E, B), B, 0, stream>>>(dstp, deg, E);
    // 3) dinv = rsqrt(deg)
    k_dinv<<<blk(N, B), B, 0, stream>>>(deg, dinv, N);
    // 4) hs1 = (x @ w1) * dinv   (WMMA f32)
    k_gemm1<<<gblk, B, 0, stream>>>(x, w1, dinv, hs1, (int)N);
    // 5) acc1[dst] += hs1[src]
    k_scat64<<<blk(E * 16, B), B, 0, stream>>>(srcp, dstp, hs1, acc1, E);
    // 6) h1 = relu(dinv*(acc1+hs1)+b1)  (in place -> acc1)
    k_fin1<<<blk(N * NHID, B), B, 0, stream>>>(acc1, hs1, dinv, b1, N);
    // 7) acc2 = 0
    k_zero<<<blk(N * NCLASS, B), B, 0, stream>>>(acc2, N * NCLASS);
    // 8) gs2 = (h1 @ w2) * dinv   (WMMA f32)
    k_gemm2<<<gblk, B, 0, stream>>>(acc1, w2, dinv, gs2, (int)N);
    // 9) acc2[dst] += gs2[src]
    k_scat32<<<blk(E * 8, B), B, 0, stream>>>(srcp, dstp, gs2, acc2, E);
    // 10) out = dinv*(acc2+gs2)+b2
    k_fin2<<<blk(N * NCLASS, B), B, 0, stream>>>(acc2, gs2, dinv, b2, out, N);
}